// BoundedSoftmax_59949153518086
// MI455X (gfx1250) — compile-verified
//
#include <hip/hip_runtime.h>

// BoundedSoftmax bound-propagation kernel for MI455X (gfx1250).
// Output: 2x (16,1024,1024) f32 coef tensors + 4x (16,1024) vectors
// = 134.5 MB -> ~5.8us HBM store floor @ 23.3 TB/s. No matmul structure ->
// WMMA inapplicable; CDNA5 async global->LDS staging is used for the inputs.
//
// This version: ONE wave32 per output row (32 threads x 32 columns).
//  - xor-butterfly wave reduction leaves row sums in every lane -> no LDS
//    reduction stage, no broadcasts, no barriers (single-wave WG => S_NOP).
//  - the wave waits on its own ASYNCcnt for its LDS staging.
//  - bias sums folded into FMA accumulators; v_rcp instead of IEEE div.

#define C_DIM   1024
#define B_DIM   16
#define THREADS 32             // exactly one wave32; each thread owns 32 cols
#define ELEMS   32
#define EPSF    1e-12f
#define TINYF   1e-9f

typedef float f4  __attribute__((ext_vector_type(4)));
typedef int   v4i __attribute__((ext_vector_type(4)));

// Typed address-space pointers for the async-to-LDS builtin
// (param type per hipcc diagnostic: "int __vector(4) __device__ *").
typedef __attribute__((address_space(1))) v4i glob_v4i;
typedef __attribute__((address_space(3))) v4i lds_v4i;

#if defined(__has_builtin)
#  if __has_builtin(__builtin_amdgcn_global_load_async_to_lds_b128)
#    define USE_ASYNC_LDS 1
#  endif
#  if __has_builtin(__builtin_amdgcn_s_wait_asynccnt)
#    define HAVE_WAIT_ASYNC_BUILTIN 1
#  endif
#  if __has_builtin(__builtin_amdgcn_rcpf)
#    define RCPF(x) __builtin_amdgcn_rcpf(x)
#  endif
#endif
#ifndef RCPF
#  define RCPF(x) (1.0f / (x))
#endif

__device__ __forceinline__ float wave_reduce_add(float v) {
#pragma unroll
  for (int off = 16; off > 0; off >>= 1)
    v += __shfl_xor(v, off, 32);
  return v;   // full sum present in every lane
}

__global__ __launch_bounds__(THREADS)
void bounded_softmax_kernel(const float* __restrict__ lower,
                            const float* __restrict__ upper,
                            float* __restrict__ out) {
  __shared__ __align__(16) float s_l[C_DIM];
  __shared__ __align__(16) float s_u[C_DIM];

  const int row = blockIdx.x;        // row = b*C + i
  const int b   = row >> 10;         // / C_DIM
  const int i   = row & (C_DIM - 1); // % C_DIM
  const int tid = threadIdx.x;
  const int j0  = tid << 5;          // 32 columns per thread

  const float* lrow = lower + (size_t)b * C_DIM;
  const float* urow = upper + (size_t)b * C_DIM;

  // ---- Stage the two 4KB input rows into LDS (async on CDNA5) ----
#if USE_ASYNC_LDS
  {
    const int st = tid << 2;   // 16B/thread/sweep; 8 sweeps cover 1024 floats
#pragma unroll
    for (int s = 0; s < 8; ++s) {
      __builtin_amdgcn_global_load_async_to_lds_b128(
          (glob_v4i*)(lrow + st + 128 * s), (lds_v4i*)(&s_l[st + 128 * s]), 0, 0);
      __builtin_amdgcn_global_load_async_to_lds_b128(
          (glob_v4i*)(urow + st + 128 * s), (lds_v4i*)(&s_u[st + 128 * s]), 0, 0);
    }
  }
#  if HAVE_WAIT_ASYNC_BUILTIN
  __builtin_amdgcn_s_wait_asynccnt(0);
#  else
  asm volatile("s_wait_asynccnt 0" ::: "memory");
#  endif
#else
  {
    const int st = tid << 2;
#pragma unroll
    for (int s = 0; s < 8; ++s) {
      *(f4*)&s_l[st + 128 * s] = *(const f4*)(lrow + st + 128 * s);
      *(f4*)&s_u[st + 128 * s] = *(const f4*)(urow + st + 128 * s);
    }
  }
#endif
  __syncthreads();   // single-wave workgroup: lowers to S_NOP

  const float li_i = s_l[i];
  const float ui_i = s_u[i];

  // ---- Pass 1: per-element relaxation coefficients + 6 row reductions ----
  // Sb_u (bias_S_upper) = S_l - sum(a_u*l_d);  Sb_l = sum(a_l) - sum(a_l*t)
  float au[ELEMS], al[ELEMS];
  float pSl = 0.f, pSu = 0.f, psAu = 0.f, psAl = 0.f;
  float pAUld = 0.f, pALt = 0.f;
#pragma unroll
  for (int c = 0; c < ELEMS / 4; ++c) {
    const f4 lv = *(const f4*)&s_l[j0 + 4 * c];
    const f4 uv = *(const f4*)&s_u[j0 + 4 * c];
#pragma unroll
    for (int k = 0; k < 4; ++k) {
      const float ld = lv[k] - ui_i;          // l_d = l_j - u_i
      const float ud = uv[k] - li_i;          // u_d = u_j - l_i
      const float el = __expf(ld);
      const float eu = __expf(ud);
      const float den = ud - ld;              // >= 0 (sum of interval widths)
      float a_u = (eu - el) * RCPF(den + EPSF);
      if (__builtin_fabsf(den) < TINYF) a_u = eu;
      const float t = 0.5f * (ld + ud);
      const float a_l = __expf(t);
      au[4 * c + k] = a_u;
      al[4 * c + k] = a_l;
      pSl += el; pSu += eu; psAu += a_u; psAl += a_l;
      pAUld = fmaf(a_u, ld, pAUld);
      pALt  = fmaf(a_l, t, pALt);
    }
  }

  // Remove the diagonal (j == i) contribution; one thread owns it.
  // On the diagonal t == 0 exactly -> a_l = 1, a_l*t = 0.
  if (tid == (i >> 5)) {
    const float ld = li_i - ui_i;
    const float ud = ui_i - li_i;
    const float el = __expf(ld);
    const float eu = __expf(ud);
    const float den = ud - ld;
    float a_u = (eu - el) * RCPF(den + EPSF);
    if (__builtin_fabsf(den) < TINYF) a_u = eu;
    pSl -= el; pSu -= eu; psAu -= a_u; psAl -= 1.0f;
    pAUld = fmaf(-a_u, ld, pAUld);
  }

  const float S_l  = wave_reduce_add(pSl);
  const float S_u  = wave_reduce_add(pSu);
  const float sAu  = wave_reduce_add(psAu);
  const float sAl  = wave_reduce_add(psAl);
  const float AUld = wave_reduce_add(pAUld);
  const float ALt  = wave_reduce_add(pALt);

  // ---- Row-scalar epilogue: every lane computes it redundantly ----
  const float bU = S_l - AUld;               // bias_S_upper
  const float bL = sAl - ALt;                // bias_S_lower
  const float g_l = RCPF(1.0f + S_l);
  const float g_u = RCPF(1.0f + S_u);
  const float dg  = S_u - S_l;
  const float tan_u = -(g_u * g_u);          // -1/(1+S_u)^2
  float m_u = (g_u - g_l) * RCPF(dg + EPSF);
  if (__builtin_fabsf(dg) < TINYF) m_u = tan_u;
  const float c_u = g_l - m_u * S_l;
  const float m_l = tan_u;
  const float c_l = g_u - m_l * S_u;
  const float dU  = -m_u * sAu;              // diagonal of upper coef
  const float dL  = -m_l * sAl;              // diagonal of lower coef

  const size_t BC  = (size_t)B_DIM * C_DIM;
  const size_t BCC = BC * C_DIM;

  if (tid == 0) {
    out[row]              = fminf(fmaxf(g_u, 0.f), 1.f);   // soft_lower
    out[BC + row]         = fminf(fmaxf(g_l, 0.f), 1.f);   // soft_upper
    out[2 * BC + 2 * BCC + row]      = m_l * bL + c_l;     // final_lower_bias
    out[2 * BC + 2 * BCC + BC + row] = m_u * bU + c_u;     // final_upper_bias
  }

  // ---- Pass 2: stream the two coef rows (non-temporal b128 stores) ----
  float* flc = out + 2 * BC + (size_t)row * C_DIM;        // final_lower_coef
  float* fuc = out + 2 * BC + BCC + (size_t)row * C_DIM;  // final_upper_coef

#pragma unroll
  for (int c = 0; c < ELEMS / 4; ++c) {
    f4 vU, vL;
#pragma unroll
    for (int k = 0; k < 4; ++k) {
      vU[k] = m_u * au[4 * c + k];
      vL[k] = m_l * al[4 * c + k];
    }
    __builtin_nontemporal_store(vL, (f4*)(flc + j0 + 4 * c));
    __builtin_nontemporal_store(vU, (f4*)(fuc + j0 + 4 * c));
  }

  // Patch the diagonal element; same-wave stores to the same address stay
  // ordered, so the owning thread overwrites its own vector-store lane.
  if (tid == (i >> 5)) {
    flc[i] = dL;
    fuc[i] = dU;
  }
}

extern "C" void kernel_launch(void* const* d_in, const int* in_sizes, int n_in,
                              void* d_out, int out_size, void* d_ws, size_t ws_size,
                              hipStream_t stream) {
  const float* lower = (const float*)d_in[0];
  const float* upper = (const float*)d_in[1];
  float* out = (float*)d_out;
  (void)in_sizes; (void)n_in; (void)out_size; (void)d_ws; (void)ws_size;

  dim3 grid(B_DIM * C_DIM);   // one wave32 workgroup per output row (b, i)
  dim3 block(THREADS);
  hipLaunchKernelGGL(bounded_softmax_kernel, grid, block, 0, stream,
                     lower, upper, out);
}